// RNN_23167053594687
// MI455X (gfx1250) — compile-verified
//
#include <hip/hip_runtime.h>
#include <math.h>

// Problem dims (fixed by the reference).
#define T_STEPS 512
#define BATCH   64
#define DIM     1024
#define LDSS    72     // padded LDS row stride in floats (bank-conflict avoidance)
#define KCA     128    // K-chunk for the big parallel GEMMs (double-buffered)
#define NCH     (DIM / KCA)
#define NB2     16     // persistent blocks in the recurrence (1024 cols / 64)

typedef __attribute__((ext_vector_type(2))) float v2f;
typedef __attribute__((ext_vector_type(8))) float v8f;
typedef int v4i __attribute__((vector_size(16)));

#define AS1 __attribute__((address_space(1)))
#define AS3 __attribute__((address_space(3)))

#if __has_builtin(__builtin_amdgcn_global_load_async_to_lds_b128)
#define HAVE_ASYNC_LDS 1
#else
#define HAVE_ASYNC_LDS 0
#endif

__device__ __forceinline__ void copy16_g2l(const float* __restrict__ g, float* l) {
#if HAVE_ASYNC_LDS
  // GLOBAL_LOAD_ASYNC_TO_LDS_B128: global -> LDS, no VGPR transit, ASYNCcnt.
  __builtin_amdgcn_global_load_async_to_lds_b128((AS1 v4i*)g, (AS3 v4i*)l, 0, 0);
#else
  *(float4*)l = *(const float4*)g;
#endif
}

__device__ __forceinline__ void wait_async0() {
#if HAVE_ASYNC_LDS
#if __has_builtin(__builtin_amdgcn_s_wait_asynccnt)
  __builtin_amdgcn_s_wait_asynccnt(0);
#else
  asm volatile("s_wait_asynccnt 0x0" ::: "memory");
#endif
#endif
}

__device__ __forceinline__ v8f wmma_f32(v2f a, v2f b, v8f c) {
  // V_WMMA_F32_16X16X4_F32 : D = A(16x4) * B(4x16) + C(16x16), all f32
  return __builtin_amdgcn_wmma_f32_16x16x4_f32(
      /*neg_a=*/false, a, /*neg_b=*/false, b,
      /*c_mod=*/(short)0, c, /*reuse_a=*/false, /*reuse_b=*/false);
}

// ---------------------------------------------------------------------------
// Parallel GEMM: C[M x 1024] = (tanh?)(A[M x 1024] @ W[1024 x 1024] + bias)
// Block tile 64(M) x 64(N); 8 waves in 4(M) x 2(N); each wave: 16x32 subtile
// = two 16x16 f32 accumulators. W streamed through a double-buffered LDS
// pipeline: async-fill chunk c+1 while WMMAs consume chunk c.
// ---------------------------------------------------------------------------
template <bool TANH>
__global__ void gemm64_kernel(const float* __restrict__ A,
                              const float* __restrict__ W,
                              const float* __restrict__ bias,
                              float* __restrict__ C) {
  extern __shared__ float Wlds[];  // 2 x (KCA x LDSS) floats = 72 KB
  const int lane = threadIdx.x & 31;
  const int hl   = lane >> 4;      // half-wave select (K pairing)
  const int l16  = lane & 15;
  const int wave = threadIdx.x >> 5;
  const int wm   = wave & 3;       // M group (16 rows each)
  const int wn   = wave >> 2;      // N group (32 cols each)

  const int mBlk = blockIdx.y * 64;
  const int nBlk = blockIdx.x * 64;
  const float* Arow = A + (size_t)(mBlk + 16 * wm + l16) * DIM;

  auto stage = [&](int c) {  // issue async fill of chunk c into buffer c&1
    const int kc = c * KCA;
    float* dst = Wlds + (c & 1) * (KCA * LDSS);
    for (int i = threadIdx.x; i < KCA * 16; i += blockDim.x) {
      int r  = i >> 4;
      int c4 = (i & 15) << 2;
      copy16_g2l(W + (size_t)(kc + r) * DIM + nBlk + c4, dst + r * LDSS + c4);
    }
  };

  v8f c0 = {};
  v8f c1 = {};

  stage(0);
  for (int c = 0; c < NCH; ++c) {
    wait_async0();        // my copies for chunk c are in LDS
    __syncthreads();      // everyone's are; everyone done reading buffer (c+1)&1
    if (c + 1 < NCH) {
      stage(c + 1);       // overlap next fill with this chunk's WMMAs
      __builtin_prefetch(Arow + (c + 1) * KCA, 0, 3);       // warm A stream
      __builtin_prefetch(Arow + (c + 1) * KCA + 64, 0, 3);
    }
    const float* lw = Wlds + (c & 1) * (KCA * LDSS);
    const int kc = c * KCA;
#pragma unroll 4
    for (int k = 0; k < KCA; k += 4) {
      v2f a = *(const v2f*)(Arow + kc + k + 2 * hl);         // A[m][k+2hl .. +1]
      const float* bp = &lw[(k + 2 * hl) * LDSS + 32 * wn + l16];
      v2f b0, b1;
      b0.x = bp[0];    b0.y = bp[LDSS];                      // B rows, cols 0..15
      b1.x = bp[16];   b1.y = bp[LDSS + 16];                 // cols 16..31
      c0 = wmma_f32(a, b0, c0);
      c1 = wmma_f32(a, b1, c1);
    }
  }

  const int col0 = nBlk + 32 * wn + l16;
#pragma unroll
  for (int r = 0; r < 8; ++r) {
    int row  = mBlk + 16 * wm + r + 8 * hl;  // C/D layout: VGPR r -> M=r / r+8
    float v0 = c0[r] + bias[col0];
    float v1 = c1[r] + bias[col0 + 16];
    if (TANH) { v0 = tanhf(v0); v1 = tanhf(v1); }
    C[(size_t)row * DIM + col0]      = v0;
    C[(size_t)row * DIM + col0 + 16] = v1;
  }
}

// ---------------------------------------------------------------------------
// Sequential recurrence: H[t] = tanh(H[t-1] @ Wh + XI[t]), H[-1] = 0.
// 16 persistent blocks; block b owns columns [64b, 64b+64) and keeps its whole
// 1024x64 Wh panel resident in 288 KB of LDS for all 512 steps. Steps are
// separated by an L2-atomic device barrier (bar[t] counts arrived blocks).
// ---------------------------------------------------------------------------
__global__ void rnn_recur_kernel(const float* __restrict__ XI,
                                 const float* __restrict__ Wh,
                                 float* __restrict__ H,
                                 int* __restrict__ bar) {
  extern __shared__ float Wlds[];  // DIM x LDSS floats (288 KB)
  const int lane = threadIdx.x & 31;
  const int hl   = lane >> 4;
  const int l16  = lane & 15;
  const int wave = threadIdx.x >> 5;
  const int wm   = wave & 3;
  const int wn   = wave >> 2;
  const int nBlk = blockIdx.x * 64;

  // Load the full Wh column panel once (async, no VGPR transit).
  for (int i = threadIdx.x; i < DIM * 16; i += blockDim.x) {
    int r  = i >> 4;
    int c4 = (i & 15) << 2;
    copy16_g2l(Wh + (size_t)r * DIM + nBlk + c4, &Wlds[r * LDSS + c4]);
  }
  wait_async0();
  __syncthreads();

  const int rowA = 16 * wm + l16;          // this lane's A row (0..63)
  const int col0 = nBlk + 32 * wn + l16;

  for (int t = 0; t < T_STEPS; ++t) {
    const float* xit = XI + (size_t)t * BATCH * DIM;
    float*       ht  = H  + (size_t)t * BATCH * DIM;

    // XI[t] is barrier-independent: pull its lines toward us while we compute.
#pragma unroll
    for (int r = 0; r < 8; ++r)
      __builtin_prefetch(xit + (size_t)(16 * wm + r + 8 * hl) * DIM + col0, 0, 3);

    v8f c0 = {};
    v8f c1 = {};
    if (t > 0) {
      const float* hprev = H + (size_t)(t - 1) * BATCH * DIM + (size_t)rowA * DIM;
#pragma unroll 4
      for (int k = 0; k < DIM; k += 4) {
        v2f a = *(const v2f*)(hprev + k + 2 * hl);
        const float* bp = &Wlds[(k + 2 * hl) * LDSS + 32 * wn + l16];
        v2f b0, b1;
        b0.x = bp[0];    b0.y = bp[LDSS];
        b1.x = bp[16];   b1.y = bp[LDSS + 16];
        c0 = wmma_f32(a, b0, c0);
        c1 = wmma_f32(a, b1, c1);
      }
    }
#pragma unroll
    for (int r = 0; r < 8; ++r) {
      int row = 16 * wm + r + 8 * hl;
      ht[(size_t)row * DIM + col0]      = tanhf(c0[r] + xit[(size_t)row * DIM + col0]);
      ht[(size_t)row * DIM + col0 + 16] = tanhf(c1[r] + xit[(size_t)row * DIM + col0 + 16]);
    }

    // Device-scope step barrier: make H[t] visible, count arrivals, spin.
    __threadfence();
    __syncthreads();
    if (threadIdx.x == 0) {
      __hip_atomic_fetch_add(&bar[t], 1, __ATOMIC_RELEASE, __HIP_MEMORY_SCOPE_AGENT);
      while (__hip_atomic_load(&bar[t], __ATOMIC_ACQUIRE, __HIP_MEMORY_SCOPE_AGENT) < NB2) {
        __builtin_amdgcn_s_sleep(4);
      }
    }
    __syncthreads();
    __builtin_amdgcn_fence(__ATOMIC_ACQUIRE, "agent");
  }
}

__global__ void init_bar_kernel(int* __restrict__ bar) {
  if (threadIdx.x < T_STEPS) bar[threadIdx.x] = 0;
}

// ---------------------------------------------------------------------------
extern "C" void kernel_launch(void* const* d_in, const int* in_sizes, int n_in,
                              void* d_out, int out_size, void* d_ws, size_t ws_size,
                              hipStream_t stream) {
  const float* x  = (const float*)d_in[0];
  const float* Wh = (const float*)d_in[1];
  const float* Wi = (const float*)d_in[2];
  const float* Wo = (const float*)d_in[3];
  const float* bh = (const float*)d_in[4];
  const float* bo = (const float*)d_in[5];
  float* out = (float*)d_out;

  const size_t NTB = (size_t)T_STEPS * BATCH * DIM;   // 33,554,432 elems
  float* XI = (float*)d_ws;        // [T,B,DIM] input projection (+bh)
  float* H  = XI + NTB;            // [T,B,DIM] hidden-state history
  int*   bar = (int*)(H + NTB);    // [T] step barrier counters

  // Reset barrier counters (stream-ordered, graph-capture safe).
  init_bar_kernel<<<1, T_STEPS, 0, stream>>>(bar);

  // Phase 1: XI = x @ Wi + bh  (parallel, WMMA, async double-buffered W)
  gemm64_kernel<false><<<dim3(DIM / 64, (T_STEPS * BATCH) / 64), 256,
                         2 * KCA * LDSS * sizeof(float), stream>>>(x, Wi, bh, XI);

  // Phase 2: sequential h-chain with Wh panel resident in LDS.
  rnn_recur_kernel<<<NB2, 256, DIM * LDSS * sizeof(float), stream>>>(XI, Wh, H, bar);

  // Phase 3: out = tanh(H @ Wo + bo)  (parallel, WMMA)
  gemm64_kernel<true><<<dim3(DIM / 64, (T_STEPS * BATCH) / 64), 256,
                        2 * KCA * LDSS * sizeof(float), stream>>>(H, Wo, bo, out);

  // h_final = H[T-1], appended after `out` in the flattened output.
  (void)hipMemcpyAsync(out + NTB, H + (size_t)(T_STEPS - 1) * BATCH * DIM,
                       (size_t)BATCH * DIM * sizeof(float),
                       hipMemcpyDeviceToDevice, stream);
}